// GCNLayer_10282151706721
// MI455X (gfx1250) — compile-verified
//
#include <hip/hip_runtime.h>

#define N_NODES 100000
#define N_EDGES 1600000
#define D_FEAT  128
#define LDS_STRIDE 132   // 128 + 4 pad floats -> conflict-free ds_load_b64

typedef __attribute__((ext_vector_type(2))) float v2f;
typedef __attribute__((ext_vector_type(8))) float v8f;

// ---------------- Kernel 1: zero the AH accumulator (lives in d_out) --------
__global__ void gcn_zero_kernel(float4* __restrict__ out, int n4) {
    int i = blockIdx.x * blockDim.x + threadIdx.x;
    int stride = gridDim.x * blockDim.x;
    for (; i < n4; i += stride)
        out[i] = make_float4(0.f, 0.f, 0.f, 0.f);
}

// ---------------- Kernel 2: SpMM scatter, one wave32 per edge ---------------
// AH[src[e], :] += val[e] * H[dst[e], :]
// lane l handles floats [4l, 4l+4). All gathers/atomics hit the 192MB L2
// since H (51.2MB) + AH (51.2MB) are co-resident.
__global__ void __launch_bounds__(256)
gcn_spmm_scatter(const float* __restrict__ H,
                 const float* __restrict__ vals,
                 const int*   __restrict__ src,
                 const int*   __restrict__ dst,
                 float*       __restrict__ AH) {
    long long tid = (long long)blockIdx.x * blockDim.x + threadIdx.x;
    int e    = (int)(tid >> 5);
    int lane = (int)(tid & 31);
    if (e >= N_EDGES) return;

    float v = vals[e];          // wave-uniform -> scalarized
    int   s = src[e];
    int   d = dst[e];

    const float4 h = *reinterpret_cast<const float4*>(H + (size_t)d * D_FEAT + lane * 4);
    float* o = AH + (size_t)s * D_FEAT + lane * 4;
    atomicAdd(o + 0, v * h.x);
    atomicAdd(o + 1, v * h.y);
    atomicAdd(o + 2, v * h.z);
    atomicAdd(o + 3, v * h.w);
}

// ---------------- Kernel 3: fused WMMA GEMM + bias + ReLU (in place) --------
// One block = 16 output rows. 8 waves; wave w owns columns [16w, 16w+16).
// A (16x128 f32) is staged into LDS first (so we can overwrite d_out rows).
__global__ void __launch_bounds__(256)
gcn_gemm_relu(const float* __restrict__ W,     // [128,128] row-major (K,N)
              const float* __restrict__ bias,  // [128]
              float*       __restrict__ Out) { // in: AH, out: relu(AH@W + b)
    __shared__ float Abuf[16 * LDS_STRIDE];

    const int row_base = blockIdx.x * 16;

    // ---- stage 16x128 A-stripe: 256 threads x 8 floats (2x float4) ----
    {
        const int   t = threadIdx.x;
        const int   g = t * 8;            // flat element in the 16x128 stripe
        const int   r = g >> 7;           // row 0..15
        const int   c = g & 127;          // col, multiple of 8
        const float4* gp = reinterpret_cast<const float4*>(
            Out + (size_t)row_base * D_FEAT + g);
        float4 x0 = gp[0];
        float4 x1 = gp[1];
        float4* lp = reinterpret_cast<float4*>(&Abuf[r * LDS_STRIDE + c]);
        lp[0] = x0;
        lp[1] = x1;
    }
    __syncthreads();

    const int lane   = threadIdx.x & 31;
    const int wave   = threadIdx.x >> 5;
    const int n_base = wave * 16;
    const int nn     = lane & 15;   // N index within tile (A: M index)
    const int half   = lane >> 4;   // lane half selects K pair / M+8

    v8f acc = {};

    // K = 128 in 32 steps of 4: D = A(16x4) x B(4x16) + C
    #pragma unroll 4
    for (int kb = 0; kb < 32; ++kb) {
        const int k0 = kb * 4 + 2 * half;
        // A frag: lane holds A[nn][k0], A[nn][k0+1]  (nn == M here)
        v2f a = *reinterpret_cast<const v2f*>(&Abuf[nn * LDS_STRIDE + k0]);
        // B frag: lane holds W[k0][n_base+nn], W[k0+1][n_base+nn]
        v2f b;
        b.x = W[(size_t)(k0)     * D_FEAT + n_base + nn];
        b.y = W[(size_t)(k0 + 1) * D_FEAT + n_base + nn];
        acc = __builtin_amdgcn_wmma_f32_16x16x4_f32(
            /*neg_a=*/false, a, /*neg_b=*/false, b,
            /*c_mod=*/(short)0, acc, /*reuse_a=*/false, /*reuse_b=*/false);
    }

    const float bv = bias[n_base + nn];
    #pragma unroll
    for (int v = 0; v < 8; ++v) {
        const int row = row_base + v + 8 * half;   // C/D: VGPR v -> M = v (+8)
        const float x = acc[v] + bv;
        Out[(size_t)row * D_FEAT + n_base + nn] = fmaxf(x, 0.f);
    }
}

// ---------------------------------------------------------------------------
extern "C" void kernel_launch(void* const* d_in, const int* in_sizes, int n_in,
                              void* d_out, int out_size, void* d_ws, size_t ws_size,
                              hipStream_t stream) {
    const float* H    = (const float*)d_in[0];   // [N_NODES, 128]
    const float* vals = (const float*)d_in[1];   // [N_EDGES]
    const float* W    = (const float*)d_in[2];   // [128, 128]
    const float* b    = (const float*)d_in[3];   // [128]
    const int*   src  = (const int*)  d_in[4];   // [N_EDGES]
    const int*   dst  = (const int*)  d_in[5];   // [N_EDGES]
    float*       out  = (float*)d_out;           // [N_NODES, 128]

    // 1) zero AH accumulator (d_out is poisoned / stale between calls)
    const int n4 = N_NODES * D_FEAT / 4;
    gcn_zero_kernel<<<4096, 256, 0, stream>>>((float4*)out, n4);

    // 2) scatter: one wave32 per edge
    const long long lanes = (long long)N_EDGES * 32;
    const int blocks = (int)((lanes + 255) / 256);
    gcn_spmm_scatter<<<blocks, 256, 0, stream>>>(H, vals, src, dst, out);

    // 3) fused WMMA projection + bias + relu, in place on d_out
    gcn_gemm_relu<<<N_NODES / 16, 256, 0, stream>>>(W, b, out);
}